// SamplewiseDTWCVAELoss_5145370821029
// MI455X (gfx1250) — compile-verified
//
#include <hip/hip_runtime.h>
#include <hip/hip_bf16.h>
#include <stdint.h>

#define NPTS   512
#define BAND   150
#define GAMMA  0.1f
#define INVG   10.0f      // 1/GAMMA
#define BIGF   1e10f
#define NB     32         // batch
#define NZ     64         // latent dim

// ---------------------------------------------------------------------------
// Kernel 1: banded soft-DTW via anti-diagonal wavefront.
// One block per pair (96 blocks: [0,32)=x·y, [32,64)=x·x, [64,96)=y·y).
// Trajectories staged into LDS with CDNA5 async global->LDS B128 loads
// (ASYNCcnt), wavefront state kept in 3 rotating LDS rows.
// ---------------------------------------------------------------------------
__global__ __launch_bounds__(256)
void sdtw_wavefront_kernel(const float* __restrict__ act,
                           const float* __restrict__ gt,
                           float* __restrict__ vout) {
    __shared__ float4 sx[NPTS];          // X trajectory, 8 KB
    __shared__ float4 sy[NPTS];          // Y trajectory, 8 KB
    __shared__ float  dbuf[3][NPTS];     // rotating wavefront rows, 6 KB

    const int tid = threadIdx.x;
    const int b   = blockIdx.x;          // 0..95

    // select source trajectories per concat group
    const float* xsrc;
    const float* ysrc;
    if (b < NB)            { xsrc = act + (size_t)b        * NPTS * 4; ysrc = gt  + (size_t)b        * NPTS * 4; }
    else if (b < 2 * NB)   { xsrc = act + (size_t)(b - NB) * NPTS * 4; ysrc = xsrc; }
    else                   { xsrc = gt  + (size_t)(b - 2*NB) * NPTS * 4; ysrc = xsrc; }

    // ---- async stage both 8KB trajectories into LDS (B128 per lane) ----
    {
        unsigned sx_lds = (unsigned)(uintptr_t)&sx[0];
        unsigned sy_lds = (unsigned)(uintptr_t)&sy[0];
        unsigned long long xa = (unsigned long long)(uintptr_t)xsrc;
        unsigned long long ya = (unsigned long long)(uintptr_t)ysrc;
        #pragma unroll
        for (int k = 0; k < 2; ++k) {
            unsigned off = (unsigned)(tid + k * 256) * 16u;
            asm volatile("global_load_async_to_lds_b128 %0, %1, %2"
                         :: "v"(sx_lds + off), "v"(off), "s"(xa) : "memory");
            asm volatile("global_load_async_to_lds_b128 %0, %1, %2"
                         :: "v"(sy_lds + off), "v"(off), "s"(ya) : "memory");
        }
        asm volatile("s_wait_asynccnt 0x0" ::: "memory");
    }

    // init wavefront rows (rows -1, -2 = BIG)
    float* flat = &dbuf[0][0];
    for (int i = tid; i < 3 * NPTS; i += 256) flat[i] = BIGF;
    __syncthreads();

    // ---- anti-diagonal scan: row p depends on rows p-1, p-2 ----
    for (int p = 0; p < 2 * NPTS - 1; ++p) {
        float* dc = dbuf[p % 3];         // row p
        float* d1 = dbuf[(p + 2) % 3];   // row p-1
        float* d2 = dbuf[(p + 1) % 3];   // row p-2
        #pragma unroll
        for (int k = 0; k < 2; ++k) {
            const int i = tid + k * 256;
            const int j = p - i;
            const int dij = (i > j) ? (i - j) : (j - i);
            float r = BIGF;
            if (j >= 0 && j < NPTS && dij <= BAND) {
                float4 xi = sx[i];
                float4 yj = sy[j];
                float c = fabsf(xi.x - yj.x) + fabsf(xi.y - yj.y)
                        + fabsf(xi.z - yj.z) + fabsf(xi.w - yj.w);
                float diag = (i == 0) ? ((j == 0) ? 0.0f : BIGF) : d2[i - 1];
                float up   = (i == 0) ? BIGF : d1[i - 1];
                float left = d1[i];
                float m = fminf(diag, fminf(up, left));
                float s = __expf((m - diag) * INVG)
                        + __expf((m - up)   * INVG)
                        + __expf((m - left) * INVG);
                r = c + m - GAMMA * __logf(s);
            }
            dc[i] = r;
        }
        __syncthreads();
    }

    if (tid == 0) vout[b] = dbuf[(2 * NPTS - 2) % 3][NPTS - 1];
}

// ---------------------------------------------------------------------------
// Kernel 2: one block. Normalized DTW, free-bits KL, transition counts,
// final 5-element loss vector.
// ---------------------------------------------------------------------------
__global__ __launch_bounds__(256)
void sdtw_finalize_kernel(const float* __restrict__ act,
                          const float* __restrict__ gt,
                          const float* __restrict__ mu,
                          const float* __restrict__ lv,
                          const float* __restrict__ pred,
                          const float* __restrict__ v,
                          float* __restrict__ out) {
    __shared__ float pGT[256], pPS[256], pKL[256];
    __shared__ float sGT[NB], sPS[NB], sKL[NB];

    const int tid = threadIdx.x;
    const int b   = tid & 31;    // batch element
    const int seg = tid >> 5;    // 8 segments

    // transition counts: channel 2 of [B,N,4]; diffs n = 1..511
    float sg = 0.0f, sp = 0.0f;
    int n0 = seg * 64; if (n0 == 0) n0 = 1;
    int n1 = seg * 64 + 64;
    for (int n = n0; n < n1; ++n) {
        float g1 = gt[((size_t)b * NPTS + n)     * 4 + 2];
        float g0 = gt[((size_t)b * NPTS + n - 1) * 4 + 2];
        sg += fabsf(g1 - g0);
        float a1 = act[((size_t)b * NPTS + n)     * 4 + 2];
        float a0 = act[((size_t)b * NPTS + n - 1) * 4 + 2];
        float t1 = 1.0f / (1.0f + __expf(-(a1 - 0.5f) * 10.0f));
        float t0 = 1.0f / (1.0f + __expf(-(a0 - 0.5f) * 10.0f));
        sp += fabsf(t1 - t0);
    }
    // KL partial: z in [seg*8, seg*8+8)
    float sk = 0.0f;
    for (int z = seg * 8; z < seg * 8 + 8; ++z) {
        float m = mu[b * NZ + z];
        float l = lv[b * NZ + z];
        sk += 1.0f + l - m * m - __expf(l);
    }
    pGT[tid] = sg; pPS[tid] = sp; pKL[tid] = sk;
    __syncthreads();

    if (seg == 0) {
        float tg = 0.0f, tp = 0.0f, tk = 0.0f;
        #pragma unroll
        for (int s2 = 0; s2 < 8; ++s2) {
            tg += pGT[s2 * 32 + b];
            tp += pPS[s2 * 32 + b];
            tk += pKL[s2 * 32 + b];
        }
        sGT[b] = tg;
        sPS[b] = tp;
        sKL[b] = fmaxf(-0.5f * tk - 0.5f, 0.0f);   // free bits = 0.5
    }
    __syncthreads();

    if (tid == 0) {
        float recon = 0.0f, kl = 0.0f, aux = 0.0f, trans = 0.0f;
        for (int bb = 0; bb < NB; ++bb) {
            float vn = v[bb] - 0.5f * (v[NB + bb] + v[2 * NB + bb]);
            recon += vn;
            kl    += sKL[bb];
            float da = pred[bb] - sGT[bb];  aux   += da * da;
            float dt = sPS[bb]  - sGT[bb];  trans += dt * dt;
        }
        recon *= (1.0f / 1024.0f);   // mean over diag(v) 32x32 matrix
        kl    *= (1.0f / 32.0f);
        aux   *= (1.0f / 32.0f);
        trans *= (1.0f / 32.0f);
        float total = 1.0f * recon + 1.0f * kl + 0.1f * aux + 0.5f * trans;
        out[0] = total;
        out[1] = recon;
        out[2] = kl;
        out[3] = aux;
        out[4] = trans;
    }
}

// ---------------------------------------------------------------------------
extern "C" void kernel_launch(void* const* d_in, const int* in_sizes, int n_in,
                              void* d_out, int out_size, void* d_ws, size_t ws_size,
                              hipStream_t stream) {
    const float* act  = (const float*)d_in[0];   // [32,512,4]
    const float* mu   = (const float*)d_in[1];   // [32,64]
    const float* lv   = (const float*)d_in[2];   // [32,64]
    const float* pred = (const float*)d_in[3];   // [32,1]
    const float* gt   = (const float*)d_in[4];   // [32,512,4]
    float* out = (float*)d_out;                  // 5 floats
    float* v   = (float*)d_ws;                   // 96 DTW values

    sdtw_wavefront_kernel<<<3 * NB, 256, 0, stream>>>(act, gt, v);
    sdtw_finalize_kernel<<<1, 256, 0, stream>>>(act, gt, mu, lv, pred, v, out);
}